// HistoryRGCN_43911745634851
// MI455X (gfx1250) — compile-verified
//
#include <hip/hip_runtime.h>
#include <hip/hip_bf16.h>
#include <math.h>

typedef __attribute__((ext_vector_type(2))) float v2f;
typedef __attribute__((ext_vector_type(8))) float v8f;

#define N0  300000
#define N1  60000
#define N2  6000
#define E1  1200000
#define E2  300000
#define D_H 128
#define D_OUT 64
#define BN_EPS 1e-5f

// ---------------------------------------------------------------- zero
__global__ void k_zero(float* __restrict__ p, long n) {
  long i = (long)blockIdx.x * blockDim.x + threadIdx.x;
  long stride = (long)gridDim.x * blockDim.x;
  for (; i < n; i += stride) p[i] = 0.0f;
}

// ------------------------------------------- zero A1 rows that will be used
__global__ void k_zeroA1(float* __restrict__ A1, const int* __restrict__ hmap) {
  int n = blockIdx.x * blockDim.x + threadIdx.x;
  int stride = gridDim.x * blockDim.x;
  for (; n < N1; n += stride) {
    if (hmap[n] < 0) {
#pragma unroll
      for (int r = 0; r < 4; ++r) {
        float* p = A1 + ((long)r * N1 + n) * 128;
        for (int c = 0; c < 128; ++c) p[c] = 0.0f;
      }
    }
  }
}

// ------------------------------------------------- layer-1 filtered scatter
// Only dst rows with hmap<0 are consumed (history overwrites the rest).
__global__ void k_scatter_f(const float* __restrict__ feat,
                            const int* __restrict__ src,
                            const int* __restrict__ dst,
                            const int* __restrict__ et,
                            const int* __restrict__ hmap,
                            float* __restrict__ acc, int nE, int nOut) {
  long total = (long)nE * 128;
  long i = (long)blockIdx.x * blockDim.x + threadIdx.x;
  long stride = (long)gridDim.x * blockDim.x;
  for (; i < total; i += stride) {
    int e = (int)(i >> 7);
    int c = (int)(i & 127);
    int d = dst[e];
    if (hmap[d] >= 0) continue;
    atomicAdd(&acc[((long)et[e] * nOut + d) * 128 + c],
              feat[(long)src[e] * 128 + c]);
  }
}

// ------------------------- history gather fill, fused BN partial statistics
// h1[n] = hbuf[hmap[n]] for history rows; accumulates per-column sum/sumsq
// for exactly those rows (computed rows add theirs in the GEMM-1 epilogue).
// Grid stride is a multiple of 128 -> each thread owns a fixed column.
__global__ void k_histfill_stats(const int* __restrict__ hmap,
                                 const float* __restrict__ hbuf,
                                 float* __restrict__ h1,
                                 float* __restrict__ csum,
                                 float* __restrict__ csq) {
  __shared__ float ls[256], lq[256];
  long total = (long)N1 * 128;
  long i = (long)blockIdx.x * blockDim.x + threadIdx.x;
  long stride = (long)gridDim.x * blockDim.x;   // multiple of 128
  float s = 0.f, q = 0.f;
  for (; i < total; i += stride) {
    int n = (int)(i >> 7);
    int c = (int)(i & 127);
    int hm = hmap[n];
    if (hm >= 0) {
      float v = hbuf[(long)hm * 128 + c];
      h1[i] = v;
      s += v; q += v * v;
    }
  }
  ls[threadIdx.x] = s; lq[threadIdx.x] = q;
  __syncthreads();
  if (threadIdx.x < 128) {   // column == threadIdx.x (256 = 2*128)
    float S = ls[threadIdx.x] + ls[threadIdx.x + 128];
    float Q = lq[threadIdx.x] + lq[threadIdx.x + 128];
    atomicAdd(&csum[threadIdx.x], S);
    atomicAdd(&csq[threadIdx.x], Q);
  }
}

// ------------------------------------------------- layer-1 WMMA GEMM
// Only for tiles containing a hmap<0 row. B staged into LDS in 64-K chunks,
// pair-swizzled + bank-padded (P1 mod 64 banks == 32 -> conflict-free b64).
#define P1 (2 * 128 + 32)
#define LDS1 (32 * P1)                  // 36 KB

__global__ void k_gemm1(const float* __restrict__ A1,   // [4][N1][128]
                        const float* __restrict__ x,    // [N0][128]
                        const float* __restrict__ W1,   // [4][128][128]
                        const float* __restrict__ Ws,   // [128][128]
                        const float* __restrict__ b1,   // [128]
                        const int*   __restrict__ hmap, // [N1]
                        const float* __restrict__ hbuf, // [HBUF][128]
                        float* __restrict__ h1,         // [N1][128] (raw h)
                        float* __restrict__ csum,
                        float* __restrict__ csq) {
  __shared__ float bs[LDS1];
  __shared__ int blockNeed;

  if (threadIdx.x == 0) blockNeed = 0;
  __syncthreads();
  int blockRow = blockIdx.x * 128;
  int rr = blockRow + threadIdx.x;
  if (threadIdx.x < 128 && rr < N1 && hmap[rr] < 0) blockNeed = 1;
  __syncthreads();
  if (!blockNeed) return;   // block-uniform

  int wave = threadIdx.x >> 5;
  int lane = threadIdx.x & 31;
  const int mtiles = N1 / 16;
  int tile = blockIdx.x * 8 + wave;
  bool valid = (tile < mtiles);
  int row0 = (valid ? tile : 0) * 16;
  bool pred = valid && (lane < 16) && (hmap[row0 + (lane & 15)] < 0);
  bool doWork = valid && (__ballot(pred) != 0ull);

  int aRow = lane & 15;
  int aK   = (lane >> 4) << 1;
  int n16  = lane & 15;

  v8f acc[8];
  v8f zero = {0.f,0.f,0.f,0.f,0.f,0.f,0.f,0.f};
#pragma unroll
  for (int t = 0; t < 8; ++t) acc[t] = zero;

  for (int seg = 0; seg < 5; ++seg) {
    const float* Ap = (seg < 4) ? (A1 + ((long)seg * N1 + row0 + aRow) * 128)
                                : (x  + (long)(row0 + aRow) * 128);
    const float* Bp = (seg < 4) ? (W1 + (long)seg * 128 * 128) : Ws;
    for (int kc = 0; kc < 128; kc += 64) {
      __syncthreads();
      for (int idx = threadIdx.x; idx < 64 * 128; idx += 256) {
        int k = idx >> 7;
        int n = idx & 127;
        bs[(k >> 1) * P1 + n * 2 + (k & 1)] = Bp[(kc + k) * 128 + n];
        if (kc == 0)  // look-ahead prefetch of the second chunk
          __builtin_prefetch(&Bp[(64 + k) * 128 + n], 0, 0);
      }
      __syncthreads();

      if (doWork) {
        v2f a = *(const v2f*)(Ap + kc + aK);
#pragma unroll 2
        for (int kr = 0; kr < 64; kr += 4) {
          v2f a_cur = a;
          if (kr < 60) a = *(const v2f*)(Ap + kc + kr + 4 + aK);
          const v2f* bpair =
              (const v2f*)(bs + ((kr >> 1) + (lane >> 4)) * P1) + n16;
          v2f bf[8];
#pragma unroll
          for (int t = 0; t < 8; ++t) bf[t] = bpair[t * 16];
#pragma unroll
          for (int t = 0; t < 8; ++t)
            acc[t] = __builtin_amdgcn_wmma_f32_16x16x4_f32(
                false, a_cur, false, bf[t], (short)0, acc[t], false, false);
        }
      }
    }
  }

  if (!doWork) return;
  // Epilogue: +bias, history overwrite, store raw h; BN stats for computed rows.
  int halfOff = (lane >> 4) << 3;
#pragma unroll
  for (int v = 0; v < 8; ++v) {
    int r = row0 + v + halfOff;
    int hm = hmap[r];
#pragma unroll
    for (int t = 0; t < 8; ++t) {
      int col = t * 16 + n16;
      float val = acc[t][v] + b1[col];
      if (hm >= 0) val = hbuf[(long)hm * 128 + col];
      else {
        atomicAdd(&csum[col], val);          // stats for non-history rows only
        atomicAdd(&csq[col], val * val);
      }
      h1[(long)r * 128 + col] = val;
    }
  }
}

// --------------------------------------- fold BN stats into scale/shift
__global__ void k_bnfinal(const float* __restrict__ csum,
                          const float* __restrict__ csq,
                          const float* __restrict__ gamma,
                          const float* __restrict__ beta,
                          float* __restrict__ scale,
                          float* __restrict__ shift) {
  int c = threadIdx.x;   // 128
  const float inv_n = 1.0f / (float)N1;
  float mean = csum[c] * inv_n;
  float var  = csq[c] * inv_n - mean * mean;
  float sc = gamma[c] * rsqrtf(var + BN_EPS);
  scale[c] = sc;
  shift[c] = beta[c] - mean * sc;
}

// --------------------------------------- normalized head rows for gemm2 self
__global__ void k_normhead(const float* __restrict__ h1,
                           const float* __restrict__ scale,
                           const float* __restrict__ shift,
                           float* __restrict__ h1n) {  // [N2][128]
  long total = (long)N2 * 128;
  long i = (long)blockIdx.x * blockDim.x + threadIdx.x;
  long stride = (long)gridDim.x * blockDim.x;
  for (; i < total; i += stride) {
    int c = (int)(i & 127);
    h1n[i] = fmaxf(0.0f, h1[i] * scale[c] + shift[c]);
  }
}

// ----------------------- layer-2 scatter with fused BN-apply + ReLU on gather
__global__ void k_scatter2n(const float* __restrict__ h1,
                            const float* __restrict__ scale,
                            const float* __restrict__ shift,
                            const int* __restrict__ src,
                            const int* __restrict__ dst,
                            const int* __restrict__ et,
                            float* __restrict__ acc) {
  long total = (long)E2 * 128;
  long i = (long)blockIdx.x * blockDim.x + threadIdx.x;
  long stride = (long)gridDim.x * blockDim.x;
  for (; i < total; i += stride) {
    int e = (int)(i >> 7);
    int c = (int)(i & 127);
    float v = fmaxf(0.0f, h1[(long)src[e] * 128 + c] * scale[c] + shift[c]);
    atomicAdd(&acc[((long)et[e] * N2 + dst[e]) * 128 + c], v);
  }
}

// ------------------------------------------------- layer-2 WMMA GEMM + log_softmax
#define P2 (2 * 64 + 32)
#define LDS2 (64 * P2)                  // 40 KB

__global__ void k_gemm2(const float* __restrict__ A2,   // [4][N2][128]
                        const float* __restrict__ h1n,  // [N2][128] normalized
                        const float* __restrict__ W2,   // [4][128][64]
                        const float* __restrict__ Ws2,  // [128][64]
                        const float* __restrict__ b2,   // [64]
                        float* __restrict__ out) {      // [N2][64]
  __shared__ float bs[LDS2];

  int wave = (int)((blockIdx.x * blockDim.x + threadIdx.x) >> 5);
  int lane = threadIdx.x & 31;
  const int mtiles = N2 / 16;
  bool valid = (wave < mtiles);
  int row0 = (valid ? wave : 0) * 16;
  int aRow = lane & 15;
  int aK   = (lane >> 4) << 1;
  int n16  = lane & 15;

  v8f acc[4];
  v8f zero = {0.f,0.f,0.f,0.f,0.f,0.f,0.f,0.f};
#pragma unroll
  for (int t = 0; t < 4; ++t) acc[t] = zero;

  for (int seg = 0; seg < 5; ++seg) {
    const float* Ap = (seg < 4) ? (A2  + ((long)seg * N2 + row0 + aRow) * 128)
                                : (h1n + (long)(row0 + aRow) * 128);
    const float* Bp = (seg < 4) ? (W2 + (long)seg * 128 * 64) : Ws2;

    __syncthreads();
    for (int idx = threadIdx.x; idx < 128 * 64; idx += 256) {
      int k = idx >> 6;
      int n = idx & 63;
      bs[(k >> 1) * P2 + n * 2 + (k & 1)] = Bp[k * 64 + n];
    }
    __syncthreads();

    if (valid) {
      v2f a = *(const v2f*)(Ap + aK);
#pragma unroll 2
      for (int k0 = 0; k0 < 128; k0 += 4) {
        v2f a_cur = a;
        if (k0 < 124) a = *(const v2f*)(Ap + k0 + 4 + aK);
        const v2f* bpair =
            (const v2f*)(bs + ((k0 >> 1) + (lane >> 4)) * P2) + n16;
        v2f bf[4];
#pragma unroll
        for (int t = 0; t < 4; ++t) bf[t] = bpair[t * 16];
#pragma unroll
        for (int t = 0; t < 4; ++t)
          acc[t] = __builtin_amdgcn_wmma_f32_16x16x4_f32(
              false, a_cur, false, bf[t], (short)0, acc[t], false, false);
      }
    }
  }

  if (!valid) return;
  int halfOff = (lane >> 4) << 3;
#pragma unroll
  for (int v = 0; v < 8; ++v) {
    int r = row0 + v + halfOff;
    float vals[4];
    float m = -1e30f;
#pragma unroll
    for (int t = 0; t < 4; ++t) {
      vals[t] = acc[t][v] + b2[t * 16 + n16];
      m = fmaxf(m, vals[t]);
    }
#pragma unroll
    for (int mask = 1; mask < 16; mask <<= 1)
      m = fmaxf(m, __shfl_xor(m, mask, 32));
    float s = 0.f;
#pragma unroll
    for (int t = 0; t < 4; ++t) s += expf(vals[t] - m);
#pragma unroll
    for (int mask = 1; mask < 16; mask <<= 1)
      s += __shfl_xor(s, mask, 32);
    float lse = m + logf(s);
#pragma unroll
    for (int t = 0; t < 4; ++t)
      out[(long)r * 64 + t * 16 + n16] = vals[t] - lse;
  }
}

// ---------------------------------------------------------------- launcher
extern "C" void kernel_launch(void* const* d_in, const int* in_sizes, int n_in,
                              void* d_out, int out_size, void* d_ws, size_t ws_size,
                              hipStream_t stream) {
  const float* x     = (const float*)d_in[0];
  const int*   src1  = (const int*)  d_in[1];
  const int*   dst1  = (const int*)  d_in[2];
  const int*   et1   = (const int*)  d_in[3];
  const int*   src2  = (const int*)  d_in[4];
  const int*   dst2  = (const int*)  d_in[5];
  const int*   et2   = (const int*)  d_in[6];
  const int*   hmap  = (const int*)  d_in[7];
  const float* hbuf  = (const float*)d_in[8];
  const float* W1    = (const float*)d_in[9];
  const float* Ws1   = (const float*)d_in[10];
  const float* b1    = (const float*)d_in[11];
  const float* gamma = (const float*)d_in[12];
  const float* beta  = (const float*)d_in[13];
  const float* W2    = (const float*)d_in[14];
  const float* Ws2   = (const float*)d_in[15];
  const float* b2    = (const float*)d_in[16];

  float* ws    = (float*)d_ws;
  float* A1    = ws;                                  // 4*N1*128
  float* A2    = A1 + (size_t)4 * N1 * 128;           // 4*N2*128
  float* sums  = A2 + (size_t)4 * N2 * 128;           // 256 (csum | csq)
  float* ss    = sums + 256;                          // 256 (scale | shift)
  float* h1    = ss + 256;                            // N1*128 (raw h)
  float* h1n   = h1 + (size_t)N1 * 128;               // N2*128 (normalized)
  float* out   = (float*)d_out;

  // zero A2 + sums + scale/shift (contiguous); A1 zeroed only where needed
  k_zero<<<512, 256, 0, stream>>>(A2, (long)4 * N2 * 128 + 512);
  k_zeroA1<<<256, 256, 0, stream>>>(A1, hmap);

  k_scatter_f<<<4096, 256, 0, stream>>>(x, src1, dst1, et1, hmap, A1, E1, N1);
  k_histfill_stats<<<512, 256, 0, stream>>>(hmap, hbuf, h1, sums, sums + 128);

  k_gemm1<<<(N1/16 + 7) / 8, 256, 0, stream>>>(A1, x, W1, Ws1, b1, hmap, hbuf,
                                               h1, sums, sums + 128);

  k_bnfinal<<<1, 128, 0, stream>>>(sums, sums + 128, gamma, beta, ss, ss + 128);
  k_normhead<<<512, 256, 0, stream>>>(h1, ss, ss + 128, h1n);

  k_scatter2n<<<2048, 256, 0, stream>>>(h1, ss, ss + 128, src2, dst2, et2, A2);

  k_gemm2<<<(N2/16 + 7) / 8, 256, 0, stream>>>(A2, h1n, W2, Ws2, b2, out);
}